// QPointQuantize_67465346285681
// MI455X (gfx1250) — compile-verified
//
#include <hip/hip_runtime.h>

// Nearest-qpoint quantization, K = 16 sorted points (f32 in, f32 out).
//
// Roofline: 8 B/element * 67.1M = 536.9 MB -> ~23 us floor at 23.3 TB/s.
// Compute: 15-level branchless midpoint select chain = 15 v_cmp (SGPR
// threshold) + 15 v_cndmask per element (~2.0e9 VALU ops total), which is
// the same order as the memory time on a wave32 part -> keep VALU minimal:
//  - qpoints via s_load_b512, midpoints via scalar-float SALU (gfx1250)
//  - 32-bit indexing (offsets < 2^28 bytes)
//  - 8 independent chains per loop body (2 x float4) for VOPD dual-issue
// Memory: B128 non-temporal load/store (no reuse, stream > L2) +
// global_prefetch_b8 for the next stride tile.

typedef __attribute__((ext_vector_type(4))) float v4f;

__device__ __forceinline__ float qsel15(float xv, const float q[16],
                                        const float m[15]) {
  float o = q[0];
#pragma unroll
  for (int k = 0; k < 15; ++k)
    o = (xv > m[k]) ? q[k + 1] : o;   // v_cmp (s,v) + v_cndmask chain
  return o;
}

__global__ __launch_bounds__(256) void QPointQuantize_kernel(
    const float* __restrict__ x, const float* __restrict__ qp,
    float* __restrict__ out, int n) {
  // Wave-uniform quantization points -> SGPRs (s_load_b512).
  float q[16];
#pragma unroll
  for (int i = 0; i < 16; ++i) q[i] = qp[i];
  // 15 decision midpoints (scalar-float ALU); x <= m[k] keeps the left
  // point, matching the reference's |x-lo| <= |x-hi| tie rule.
  float m[15];
#pragma unroll
  for (int i = 0; i < 15; ++i) m[i] = 0.5f * (q[i] + q[i + 1]);

  const int n4     = n >> 2;
  const int tid    = blockIdx.x * blockDim.x + threadIdx.x;
  const int stride = gridDim.x * blockDim.x;
  const v4f* __restrict__ x4 = (const v4f*)x;
  v4f* __restrict__ o4       = (v4f*)out;

  // 2-way interleaved grid-stride: two B128 loads in flight per iteration.
  for (int i = tid; i < n4; i += 2 * stride) {
    const int j      = i + stride;
    const bool has_j = (j < n4);

    __builtin_prefetch(x4 + i + 2 * stride, 0, 0);  // global_prefetch_b8

    v4f a = __builtin_nontemporal_load(x4 + i);
    v4f b;
    if (has_j) {
      __builtin_prefetch(x4 + j + 2 * stride, 0, 0);
      b = __builtin_nontemporal_load(x4 + j);
    }

    v4f ra;
    ra.x = qsel15(a.x, q, m);
    ra.y = qsel15(a.y, q, m);
    ra.z = qsel15(a.z, q, m);
    ra.w = qsel15(a.w, q, m);
    __builtin_nontemporal_store(ra, o4 + i);

    if (has_j) {
      v4f rb;
      rb.x = qsel15(b.x, q, m);
      rb.y = qsel15(b.y, q, m);
      rb.z = qsel15(b.z, q, m);
      rb.w = qsel15(b.w, q, m);
      __builtin_nontemporal_store(rb, o4 + j);
    }
  }

  // Scalar tail for n not divisible by 4 (zero here; kept generic).
  const int tail = n & 3;
  if (tid < tail) {
    const int i = (n4 << 2) + tid;
    out[i] = qsel15(x[i], q, m);
  }
}

extern "C" void kernel_launch(void* const* d_in, const int* in_sizes, int n_in,
                              void* d_out, int out_size, void* d_ws, size_t ws_size,
                              hipStream_t stream) {
  (void)n_in; (void)out_size; (void)d_ws; (void)ws_size;
  const float* x  = (const float*)d_in[0];   // (64,1024,1024) f32
  const float* qp = (const float*)d_in[1];   // (16,) f32 sorted
  float* out      = (float*)d_out;           // f32, same shape as x

  const int n       = in_sizes[0];
  const int threads = 256;
  const int n4      = n >> 2;
  // ~8 float4s per thread (4 iterations of the 2-way body) to amortize
  // the qpoint/midpoint setup while keeping tens of thousands of waves
  // resident for HBM latency hiding.
  long long want = ((long long)n4 + (long long)threads * 8 - 1) /
                   ((long long)threads * 8);
  if (want < 1) want = 1;
  if (want > 1048576) want = 1048576;
  const int blocks = (int)want;

  hipLaunchKernelGGL(QPointQuantize_kernel, dim3(blocks), dim3(threads), 0,
                     stream, x, qp, out, n);
}